// UniformMatcher_32100585571109
// MI455X (gfx1250) — compile-verified
//
#include <hip/hip_runtime.h>
#include <cstdint>
#include <cstddef>

typedef float v2f __attribute__((ext_vector_type(2)));
typedef float v8f __attribute__((ext_vector_type(8)));

#define UM_B  8
#define UM_Q  32768
#define UM_T  64
#define UM_MT 4

// One wave per (batch b, source src, target t). Block = 256 threads = 8 waves
// = 8 targets. grid = (T/8 target-groups, 2 sources, B batches) = 128 blocks.
// Fused: never materializes the [B,Q,T] cost matrix (saves ~536 MB of HBM
// traffic vs the reference dataflow); streams 8 MB of boxes through L2.
__global__ __launch_bounds__(256)
void uniform_matcher_topk_kernel(const float* __restrict__ pred,
                                 const float* __restrict__ anch,
                                 const float* __restrict__ gt,
                                 long long* __restrict__ out)
{
    __shared__ float4 s_gt[UM_T];                    // converted GT boxes (this b)
    __shared__ unsigned long long s_cand[8][128];    // per-wave candidate pool

    const int tid  = threadIdx.x;
    const int lane = tid & 31;          // wave32
    const int wv   = tid >> 5;
    const int tg   = blockIdx.x;        // target group (0..7)
    const int src  = blockIdx.y;        // 0 = pred_boxes, 1 = anchors
    const int b    = blockIdx.z;        // batch

    // Convert the 64 GT boxes for this batch to cxcywh once, into LDS.
    if (tid < UM_T) {
        const float* gp = gt + ((size_t)b * UM_T + tid) * 4;
        float x0 = gp[0], y0 = gp[1], x1 = gp[2], y1 = gp[3];
        s_gt[tid] = make_float4((x0 + x1) * 0.5f, (y0 + y1) * 0.5f,
                                x1 - x0, y1 - y0);
    }
    __syncthreads();

    const int t = tg * 8 + wv;          // this wave's target
    const float4 g = s_gt[t];

    const float* srcp = (src == 0 ? pred : anch) + (size_t)b * UM_Q * 4;

    // Running smallest-4 as sorted u64 keys: (cost_bits << 32) | query_idx.
    // cost >= 0 so IEEE bits are order-preserving; low word gives the
    // smallest-index tiebreak that lax.top_k uses.
    unsigned long long k0 = ~0ull, k1 = ~0ull, k2 = ~0ull, k3 = ~0ull;

#if __has_builtin(__builtin_amdgcn_wmma_f32_16x16x4_f32)
    // xyxy -> cxcywh is the 4x4 linear map
    //   T = [[.5,0,.5,0],[0,.5,0,.5],[-1,0,1,0],[0,-1,0,1]]
    // Done with V_WMMA_F32_16X16X4_F32: D = A(16x4 transform) x B(4x16 coords).
    // A layout: lane L<16 -> row L cols {0,1}; lane 16+L -> row L cols {2,3}.
    // WMMA #1 puts T in rows 0-3  -> results land in lanes 0-15  (D rows 0-3).
    // WMMA #2 puts T in rows 8-11 -> results land in lanes 16-31 (D rows 8-11).
    // Chaining through the C accumulator merges both halves: every lane ends
    // with its own query's converted coords in D[0..3]. Zero shuffles needed.
    const int l  = lane & 15;
    const int hi = lane >> 4;
    v2f a1 = {0.f, 0.f}, a2 = {0.f, 0.f};
    {
        const float s = hi ? 1.f : -1.f;   // w/h rows: -x0+x1 , -y0+y1
        if (l == 0)  a1[0] = 0.5f;
        if (l == 1)  a1[1] = 0.5f;
        if (l == 2)  a1[0] = s;
        if (l == 3)  a1[1] = s;
        if (l == 8)  a2[0] = 0.5f;
        if (l == 9)  a2[1] = 0.5f;
        if (l == 10) a2[0] = s;
        if (l == 11) a2[1] = s;
    }
    // B layout: lane n<16 supplies B[0][n],B[1][n] (coords 0,1 of query n);
    // lane 16+n supplies B[2][n],B[3][n] (coords 2,3). Per-lane b64 load.
    const float* base = srcp + (size_t)l * 4 + (size_t)hi * 2;

    for (int q0 = 0; q0 < UM_Q; q0 += 32) {
        const float* p = base + (size_t)q0 * 4;
        v2f b1 = *(const v2f*)p;            // queries q0 .. q0+15
        v2f b2 = *(const v2f*)(p + 64);     // queries q0+16 .. q0+31
        __builtin_prefetch(p + 2048, 0, 0); // stream-ahead into WGP/L2

        v8f c = {0.f, 0.f, 0.f, 0.f, 0.f, 0.f, 0.f, 0.f};
        c = __builtin_amdgcn_wmma_f32_16x16x4_f32(false, a1, false, b1,
                                                  (short)0, c, false, false);
        c = __builtin_amdgcn_wmma_f32_16x16x4_f32(false, a2, false, b2,
                                                  (short)0, c, false, false);
        // lane owns query q0+lane; converted coords in c[0..3]
        float cost = fabsf(c[0] - g.x) + fabsf(c[1] - g.y) +
                     fabsf(c[2] - g.z) + fabsf(c[3] - g.w);
        unsigned long long key =
            ((unsigned long long)__float_as_uint(cost) << 32) |
            (unsigned)(q0 + lane);
        if (key < k3) {                      // execz-skippable once warm
            k3 = key;
            if (k3 < k2) { unsigned long long x = k2; k2 = k3; k3 = x; }
            if (k2 < k1) { unsigned long long x = k1; k1 = k2; k2 = x; }
            if (k1 < k0) { unsigned long long x = k0; k0 = k1; k1 = x; }
        }
    }
#else
    // VALU fallback (6 ops/convert) if the f32 16x16x4 WMMA builtin is absent.
    for (int q0 = 0; q0 < UM_Q; q0 += 32) {
        const int q = q0 + lane;
        float4 v = *(const float4*)(srcp + (size_t)q * 4);
        float cx = (v.x + v.z) * 0.5f, cy = (v.y + v.w) * 0.5f;
        float w  = v.z - v.x,          h  = v.w - v.y;
        float cost = fabsf(cx - g.x) + fabsf(cy - g.y) +
                     fabsf(w - g.z) + fabsf(h - g.w);
        unsigned long long key =
            ((unsigned long long)__float_as_uint(cost) << 32) | (unsigned)q;
        if (key < k3) {
            k3 = key;
            if (k3 < k2) { unsigned long long x = k2; k2 = k3; k3 = x; }
            if (k2 < k1) { unsigned long long x = k1; k1 = k2; k2 = x; }
            if (k1 < k0) { unsigned long long x = k0; k0 = k1; k1 = x; }
        }
    }
#endif

    // Wave-level 128 -> 4 selection through this wave's LDS slice.
    unsigned long long* cd = s_cand[wv];
    cd[lane * 4 + 0] = k0;
    cd[lane * 4 + 1] = k1;
    cd[lane * 4 + 2] = k2;
    cd[lane * 4 + 3] = k3;
    asm volatile("s_wait_dscnt 0" ::: "memory");   // same-wave DS visibility

    if (lane == 0) {
        // idx_i: [B, T*2*MT]; slot (t*8 + src*4 + r). idx_j pattern = r.
        long long* oi = out + (size_t)b * (UM_T * 2 * UM_MT)
                            + (size_t)t * (2 * UM_MT) + src * UM_MT;
        long long* oj = oi + (size_t)UM_B * UM_T * 2 * UM_MT;
        for (int r = 0; r < UM_MT; ++r) {
            unsigned long long best = cd[0];
            int bp = 0;
            for (int i = 1; i < 128; ++i) {
                unsigned long long v = cd[i];
                if (v < best) { best = v; bp = i; }
            }
            cd[bp] = ~0ull;
            oi[r] = (long long)(best & 0xffffffffull);  // query index
            oj[r] = r;                                  // 0,1,2,3 pattern
        }
    }
}

extern "C" void kernel_launch(void* const* d_in, const int* in_sizes, int n_in,
                              void* d_out, int out_size, void* d_ws, size_t ws_size,
                              hipStream_t stream) {
    (void)in_sizes; (void)n_in; (void)out_size; (void)d_ws; (void)ws_size;
    const float* pred = (const float*)d_in[0];
    const float* anch = (const float*)d_in[1];
    const float* gt   = (const float*)d_in[2];
    long long*   out  = (long long*)d_out;   // int64 outputs (idx_i ++ idx_j)

    dim3 grid(UM_T / 8, 2, UM_B);   // (target-groups, src, batch)
    dim3 block(256);                // 8 wave32 waves
    hipLaunchKernelGGL(uniform_matcher_topk_kernel, grid, block, 0, stream,
                       pred, anch, gt, out);
}